// GlobalAttention_41635412968032
// MI455X (gfx1250) — compile-verified
//
#include <hip/hip_runtime.h>
#include <hip/hip_bf16.h>
#include <math.h>

typedef __attribute__((ext_vector_type(2))) float v2f;
typedef __attribute__((ext_vector_type(8))) float v8f;
typedef int vint4 __attribute__((vector_size(4 * sizeof(int))));  // matches builtin param

#define S_LEN 96
#define T_LEN 96
#define BATCH 16
#define ENC 512
#define DEC 512
#define ROWS (T_LEN * BATCH)   // 1536 rows for both proj GEMMs
#define WA_LD 1024             // Wa is (512, 1024) row-major

// ---------------------------------------------------------------------------
// Async global->LDS helpers (ASYNCcnt-tracked). The gfx1250 builtin takes
// (v4i addrspace(1)*, v4i addrspace(3)*, imm offset, imm cpol) per the clang
// diagnostic from the previous probe round.
// ---------------------------------------------------------------------------
__device__ __forceinline__ void async_copy16(const float* __restrict__ g, float* l) {
#if __has_builtin(__builtin_amdgcn_global_load_async_to_lds_b128)
    __builtin_amdgcn_global_load_async_to_lds_b128(
        (__attribute__((address_space(1))) vint4*)g,
        (__attribute__((address_space(3))) vint4*)l, /*offset=*/0, /*cpol=*/0);
#else
    unsigned lds_off = (unsigned)(unsigned long long)l;  // LDS addr = low 32 bits
    asm volatile("global_load_async_to_lds_b128 %0, %1, off"
                 :: "v"(lds_off), "v"(g) : "memory");
#endif
}

template <int N>
__device__ __forceinline__ void wait_asynccnt() {
#if __has_builtin(__builtin_amdgcn_s_wait_asynccnt)
    __builtin_amdgcn_s_wait_asynccnt((unsigned short)N);
#else
    asm volatile("s_wait_asynccnt %0" :: "i"(N) : "memory");
#endif
}

// Stage one 512-float row (2 KB) into LDS: 4 x b128 per lane.
__device__ __forceinline__ void stage_row_async(const float* __restrict__ g,
                                                float* l, int lane) {
    #pragma unroll
    for (int c = 0; c < 4; ++c) {
        const int off = (lane + 32 * c) * 4;   // 16B-aligned float offset
        async_copy16(g + off, l + off);
    }
}

// ---------------------------------------------------------------------------
// Kernel 1: proj[r, o] = sum_d X[r, d] * Wa[o, colOff + d]
//   X: (1536, 512) row-major; out: (1536, 512) row-major.
//   One 16x16 output tile per wave, V_WMMA_F32_16X16X4_F32, K=512 in steps of 4.
// ---------------------------------------------------------------------------
__global__ __launch_bounds__(128)
void proj_gemm_wmma(const float* __restrict__ X, const float* __restrict__ Wa,
                    int colOff, float* __restrict__ out) {
    const int wave = threadIdx.x >> 5;
    const int lane = threadIdx.x & 31;
    const int tile = blockIdx.x * 4 + wave;        // 0..3071
    const int tm = (tile >> 5) << 4;               // row-tile base (96 tiles of 16)
    const int tn = (tile & 31) << 4;               // col-tile base (32 tiles of 16)

    const int lr = lane & 15;                      // 0..15
    const int hi = lane >> 4;                      // 0 or 1 (selects K pair {0,1} vs {2,3})

    // A fragment rows and B fragment rows are fixed for the whole K loop.
    const float* arow = X  + (size_t)(tm + lr) * DEC + hi * 2;            // A[tm+lr, k0+hi*2..]
    const float* brow = Wa + (size_t)(tn + lr) * WA_LD + colOff + hi * 2; // W[tn+lr, k0+hi*2..]

    v8f acc = {};
    #pragma unroll 4
    for (int k0 = 0; k0 < DEC; k0 += 4) {
        v2f a, b;
        a.x = arow[k0 + 0];
        a.y = arow[k0 + 1];
        b.x = brow[k0 + 0];
        b.y = brow[k0 + 1];
        acc = __builtin_amdgcn_wmma_f32_16x16x4_f32(
            /*neg_a=*/false, a, /*neg_b=*/false, b,
            /*c_mod=*/(short)0, acc, /*reuse_a=*/false, /*reuse_b=*/false);
    }

    // D layout: VGPR v, lanes 0-15 -> row tm+v, lanes 16-31 -> row tm+v+8; col = tn+lr.
    float* orow = out + (size_t)(tm + hi * 8) * DEC + tn + lr;
    #pragma unroll
    for (int v = 0; v < 8; ++v)
        orow[(size_t)v * DEC] = acc[v];
}

// ---------------------------------------------------------------------------
// Kernel 2: fused scores + masked softmax + context, one block per (t, b).
//   scores[s] = sum_o Va[o] * tanh(pt[t,b,o] + ps[s,b,o])
//   attn = softmax_s(scores, mask s < len[b])
//   context[t,b,e] = sum_s attn[s] * src[s,b,e]
// Each of 8 waves owns 12 s-rows, double-buffering proj_s rows through LDS
// with ASYNCcnt-tracked async loads overlapped against the tanh reduction.
// ---------------------------------------------------------------------------
__global__ __launch_bounds__(256)
void attn_fused(const float* __restrict__ proj_t, const float* __restrict__ proj_s,
                const float* __restrict__ src, const int* __restrict__ lens,
                const float* __restrict__ Va, float* __restrict__ out) {
    __shared__ __align__(16) float s_ps[8][2][DEC];   // per-wave double buffers (32 KB)
    __shared__ __align__(16) float s_pt[DEC];
    __shared__ __align__(16) float s_va[DEC];
    __shared__ float s_sc[S_LEN];

    const int t = blockIdx.x;
    const int b = blockIdx.y;
    const int tid = threadIdx.x;
    const int wave = tid >> 5;
    const int lane = tid & 31;

    // Stage proj_t[t,b,:] and Va into LDS.
    const float* ptrow = proj_t + ((size_t)t * BATCH + b) * DEC;
    s_pt[tid]       = ptrow[tid];
    s_pt[tid + 256] = ptrow[tid + 256];
    s_va[tid]       = Va[tid];
    s_va[tid + 256] = Va[tid + 256];
    __syncthreads();

    // Prefetch this wave's first proj_s row (s = wave).
    stage_row_async(proj_s + ((size_t)wave * BATCH + b) * DEC, &s_ps[wave][0][0], lane);

    #pragma unroll 1
    for (int i = 0; i < 12; ++i) {                 // s = wave + 8*i
        const int s   = wave + 8 * i;
        const int cur = i & 1;

        if (i + 1 < 12) {
            // Issue next row into the other buffer, then wait for current row
            // (ASYNCcnt in-order: <=4 outstanding means the first 4 landed).
            stage_row_async(proj_s + ((size_t)(s + 8) * BATCH + b) * DEC,
                            &s_ps[wave][cur ^ 1][0], lane);
            wait_asynccnt<4>();
        } else {
            wait_asynccnt<0>();
        }

        const float* psrow = &s_ps[wave][cur][0];
        float part = 0.0f;
        #pragma unroll
        for (int o = lane; o < DEC; o += 32)
            part += s_va[o] * tanhf(s_pt[o] + psrow[o]);
        #pragma unroll
        for (int off = 16; off > 0; off >>= 1)
            part += __shfl_down(part, off, 32);
        if (lane == 0) s_sc[s] = part;
    }
    __syncthreads();

    // Masked softmax over s (96 values), wave 0. Max is over ALL scores
    // (reference takes max before masking); masked lanes excluded from the sum.
    if (wave == 0) {
        const int len = lens[b];
        float m = -3.4e38f;
        for (int j = lane; j < S_LEN; j += 32)
            m = fmaxf(m, s_sc[j]);
        #pragma unroll
        for (int off = 16; off > 0; off >>= 1)
            m = fmaxf(m, __shfl_xor(m, off, 32));

        const int j0 = lane, j1 = lane + 32, j2 = lane + 64;
        const float e0 = (j0 < len) ? __expf(s_sc[j0] - m) : 0.0f;
        const float e1 = (j1 < len) ? __expf(s_sc[j1] - m) : 0.0f;
        const float e2 = (j2 < len) ? __expf(s_sc[j2] - m) : 0.0f;
        float sum = e0 + e1 + e2;
        #pragma unroll
        for (int off = 16; off > 0; off >>= 1)
            sum += __shfl_xor(sum, off, 32);
        const float inv = 1.0f / sum;
        s_sc[j0] = e0 * inv;
        s_sc[j1] = e1 * inv;
        s_sc[j2] = e2 * inv;
    }
    __syncthreads();

    // Context GEMV: each thread owns cols tid and tid+256.
    float acc0 = 0.0f, acc1 = 0.0f;
    #pragma unroll 4
    for (int s = 0; s < S_LEN; ++s) {
        const float a = s_sc[s];
        const float* srow = src + ((size_t)s * BATCH + b) * ENC;
        acc0 = fmaf(a, srow[tid],       acc0);
        acc1 = fmaf(a, srow[tid + 256], acc1);
    }
    float* orow = out + ((size_t)t * BATCH + b) * ENC;
    orow[tid]       = acc0;
    orow[tid + 256] = acc1;
}

// ---------------------------------------------------------------------------
// Launch: inputs are (h_t, src_encodings, src_lengths, Wa, Va) per setup_inputs.
// ---------------------------------------------------------------------------
extern "C" void kernel_launch(void* const* d_in, const int* in_sizes, int n_in,
                              void* d_out, int out_size, void* d_ws, size_t ws_size,
                              hipStream_t stream) {
    const float* h_t  = (const float*)d_in[0];   // (96,16,512)  == (1536,512)
    const float* src  = (const float*)d_in[1];   // (96,16,512)  == (1536,512)
    const int*   lens = (const int*)  d_in[2];   // (16,)
    const float* Wa   = (const float*)d_in[3];   // (512,1024)
    const float* Va   = (const float*)d_in[4];   // (512,)

    float* proj_t = (float*)d_ws;                // 1536*512 floats = 3 MB
    float* proj_s = proj_t + (size_t)ROWS * DEC; // 3 MB more

    // 3072 tiles (96 row-tiles x 32 col-tiles), 4 waves/block -> 768 blocks.
    proj_gemm_wmma<<<768, 128, 0, stream>>>(h_t, Wa, /*colOff=*/0,   proj_t);
    proj_gemm_wmma<<<768, 128, 0, stream>>>(src, Wa, /*colOff=*/DEC, proj_s);

    attn_fused<<<dim3(T_LEN, BATCH), 256, 0, stream>>>(proj_t, proj_s, src, lens, Va,
                                                       (float*)d_out);
}